// Task1SupervisedModel_15539191677323
// MI455X (gfx1250) — compile-verified
//
#include <hip/hip_runtime.h>
#include <hip/hip_bf16.h>
#include <math.h>

#define DIM 128

typedef __attribute__((ext_vector_type(2))) float v2f;
typedef __attribute__((ext_vector_type(8))) float v8f;

// ---------------------------------------------------------------------------
// GEMM: Y[M x 128] = X[M x 128] * W[128 x 128], row-major, full f32 via
// v_wmma_f32_16x16x4_f32.  One wave per 16-row tile; 8 column tiles per wave.
// W cached in LDS (64 KB) and reused by all 8 waves / all row tiles.
// ---------------------------------------------------------------------------
__global__ __launch_bounds__(256) void gemm_wmma_f32(
    const float* __restrict__ X, const float* __restrict__ W,
    float* __restrict__ Y, int M)
{
    __shared__ float sW[DIM * DIM];               // 64 KB
    for (int i = threadIdx.x; i < DIM * DIM / 4; i += blockDim.x)
        ((float4*)sW)[i] = ((const float4*)W)[i];
    __syncthreads();

    const int lane    = threadIdx.x & 31;
    const int wave    = threadIdx.x >> 5;
    const int halfsel = lane >> 4;                // 0: lanes 0-15, 1: lanes 16-31
    const int l16     = lane & 15;
    const int tiles   = M >> 4;

    for (int t = blockIdx.x * 8 + wave; t < tiles; t += gridDim.x * 8) {
        const float* xrow = X + (size_t)(t * 16 + l16) * DIM;  // A row of this lane
        v8f acc[8] = {};
        for (int k = 0; k < DIM; k += 4) {
            const int kb = k + 2 * halfsel;       // A: K=0,1 (lo lanes) / K=2,3 (hi)
            const float2 av = *(const float2*)(xrow + kb);
            v2f a; a.x = av.x; a.y = av.y;
            #pragma unroll
            for (int nb = 0; nb < 8; ++nb) {      // B: K rows striped like A halves
                v2f b;
                b.x = sW[(size_t)kb * DIM + nb * 16 + l16];
                b.y = sW[(size_t)(kb + 1) * DIM + nb * 16 + l16];
                acc[nb] = __builtin_amdgcn_wmma_f32_16x16x4_f32(
                    false, a, false, b, (short)0, acc[nb], false, false);
            }
        }
        // C/D layout: VGPR r -> M = r + 8*halfsel, N = l16 (+16*nb)
        float* yt = Y + (size_t)t * 16 * DIM;
        #pragma unroll
        for (int nb = 0; nb < 8; ++nb)
            #pragma unroll
            for (int r = 0; r < 8; ++r)
                yt[(size_t)(r + 8 * halfsel) * DIM + nb * 16 + l16] = acc[nb][r];
    }
}

// ---------------------------------------------------------------------------
// Degree accumulation: one thread per incidence.
// ---------------------------------------------------------------------------
__global__ void degrees_kernel(const int* __restrict__ nidx,
                               const int* __restrict__ eidx,
                               float* Dn, float* De, int nnz)
{
    int i = blockIdx.x * blockDim.x + threadIdx.x;
    if (i < nnz) {
        atomicAdd(&Dn[nidx[i]], 1.0f);
        atomicAdd(&De[eidx[i]], 1.0f);
    }
}

__global__ void invert_kernel(float* v, int n)
{
    int i = blockIdx.x * blockDim.x + threadIdx.x;
    if (i < n) { float d = v[i]; v[i] = d > 0.0f ? 1.0f / d : 0.0f; }
}

// ---------------------------------------------------------------------------
// Segmented scatter-mean message pass: one wave per incidence, float4 lanes.
// dst[sidx[i]] += scale[sidx[i]] * src[gidx[i]]
// ---------------------------------------------------------------------------
__global__ void scatter_kernel(const float* __restrict__ src,
                               float* __restrict__ dst,
                               const int* __restrict__ gidx,
                               const int* __restrict__ sidx,
                               const float* __restrict__ scale, int nnz)
{
    int inc  = (blockIdx.x * blockDim.x + threadIdx.x) >> 5;
    int lane = threadIdx.x & 31;
    if (inc >= nnz) return;
    int g = gidx[inc];
    int s = sidx[inc];
    float sc = scale[s];
    float4 v = *(const float4*)(src + (size_t)g * DIM + lane * 4);
    float* d = dst + (size_t)s * DIM + lane * 4;
    atomicAdd(d + 0, v.x * sc);
    atomicAdd(d + 1, v.y * sc);
    atomicAdd(d + 2, v.z * sc);
    atomicAdd(d + 3, v.w * sc);
}

// ---------------------------------------------------------------------------
// Fused bias + PReLU (in place): buf[r,c] = prelu(buf[r,c] + bias[c], a)
// ---------------------------------------------------------------------------
__global__ void bias_prelu_kernel(float* buf, const float* __restrict__ bias,
                                  const float* __restrict__ pa, size_t n)
{
    size_t i = (size_t)blockIdx.x * blockDim.x + threadIdx.x;
    if (i >= n) return;
    float a = *pa;
    float v = buf[i] + bias[i & (DIM - 1)];
    buf[i] = v >= 0.0f ? v : a * v;
}

// ---------------------------------------------------------------------------
// Query pooling: Ze[Emap[i]] += x[Vmap[i]], cnt[Emap[i]] += 1
// ---------------------------------------------------------------------------
__global__ void qscatter_kernel(const float* __restrict__ x,
                                const int* __restrict__ vmap,
                                const int* __restrict__ emap,
                                float* __restrict__ ze, float* __restrict__ cnt,
                                int lq)
{
    int inc  = (blockIdx.x * blockDim.x + threadIdx.x) >> 5;
    int lane = threadIdx.x & 31;
    if (inc >= lq) return;
    int v = vmap[inc];
    int e = emap[inc];
    float4 f = *(const float4*)(x + (size_t)v * DIM + lane * 4);
    float* d = ze + (size_t)e * DIM + lane * 4;
    atomicAdd(d + 0, f.x);
    atomicAdd(d + 1, f.y);
    atomicAdd(d + 2, f.z);
    atomicAdd(d + 3, f.w);
    if (lane == 0) atomicAdd(&cnt[e], 1.0f);
}

// ---------------------------------------------------------------------------
// pred[q] = sigmoid( dot(Ze1[q],Ze2[q]) / (max(c1,1)*max(c2,1)) )
// one wave per query, shfl reduction.
// ---------------------------------------------------------------------------
__global__ void pred_kernel(const float* __restrict__ ze1,
                            const float* __restrict__ c1,
                            const float* __restrict__ ze2,
                            const float* __restrict__ c2,
                            float* __restrict__ out, int nq)
{
    int q    = (blockIdx.x * blockDim.x + threadIdx.x) >> 5;
    int lane = threadIdx.x & 31;
    if (q >= nq) return;
    float4 a = *(const float4*)(ze1 + (size_t)q * DIM + lane * 4);
    float4 b = *(const float4*)(ze2 + (size_t)q * DIM + lane * 4);
    float d = a.x * b.x + a.y * b.y + a.z * b.z + a.w * b.w;
    #pragma unroll
    for (int off = 16; off > 0; off >>= 1)
        d += __shfl_xor(d, off, 32);
    if (lane == 0) {
        float k1 = fmaxf(c1[q], 1.0f);
        float k2 = fmaxf(c2[q], 1.0f);
        float z = d / (k1 * k2);
        out[q] = 1.0f / (1.0f + expf(-z));
    }
}

// ---------------------------------------------------------------------------
extern "C" void kernel_launch(void* const* d_in, const int* in_sizes, int n_in,
                              void* d_out, int out_size, void* d_ws, size_t ws_size,
                              hipStream_t stream)
{
    const float* x     = (const float*)d_in[0];
    const int*   hei   = (const int*)d_in[1];
    const int*   Vmap1 = (const int*)d_in[4];
    const int*   Vmap2 = (const int*)d_in[5];
    const int*   Emap1 = (const int*)d_in[6];
    const int*   Emap2 = (const int*)d_in[7];
    const float* W1n2e = (const float*)d_in[9];
    const float* b1n2e = (const float*)d_in[10];
    const float* W1e2n = (const float*)d_in[11];
    const float* b1e2n = (const float*)d_in[12];
    const float* W2n2e = (const float*)d_in[13];
    const float* b2n2e = (const float*)d_in[14];
    const float* W2e2n = (const float*)d_in[15];
    const float* b2e2n = (const float*)d_in[16];
    const float* pa    = (const float*)d_in[17];
    float* out = (float*)d_out;

    const int N   = in_sizes[0] / DIM;   // 100000 nodes
    const int NNZ = in_sizes[1] / 2;     // 1000000 incidences
    const int NE  = 50000;               // num_edges (reference constant)
    const int NQ  = out_size;            // 100000 queries
    const int LQ  = in_sizes[4];         // 500000 query incidences

    const int* nidx = hei;
    const int* eidx = hei + NNZ;

    // workspace carve-up (floats)
    float* fA  = (float*)d_ws;                // N*128   (h / h2, reused)
    float* fB  = fA  + (size_t)N  * DIM;      // N*128   (layer output x)
    float* fE  = fB  + (size_t)N  * DIM;      // NE*128  (edge features e)
    float* Ze1 = fE  + (size_t)NE * DIM;      // NQ*128
    float* Ze2 = Ze1 + (size_t)NQ * DIM;      // NQ*128
    float* DnI = Ze2 + (size_t)NQ * DIM;      // N
    float* DeI = DnI + N;                     // NE
    float* c1  = DeI + NE;                    // NQ
    float* c2  = c1  + NQ;                    // NQ

    const int TB = 256;
    auto cdiv = [](long long a, long long b) { return (int)((a + b - 1) / b); };
    const int gemmBlkN  = cdiv(N  / 16, 8);
    const int gemmBlkE  = cdiv(NE / 16, 8);
    const int scatGrid  = cdiv((long long)NNZ * 32, TB);
    const int qscatGrid = cdiv((long long)LQ  * 32, TB);

    // ---- degrees ----
    hipMemsetAsync(DnI, 0, (size_t)N  * sizeof(float), stream);
    hipMemsetAsync(DeI, 0, (size_t)NE * sizeof(float), stream);
    degrees_kernel<<<cdiv(NNZ, TB), TB, 0, stream>>>(nidx, eidx, DnI, DeI, NNZ);
    invert_kernel<<<cdiv(N,  TB), TB, 0, stream>>>(DnI, N);
    invert_kernel<<<cdiv(NE, TB), TB, 0, stream>>>(DeI, NE);

    // ---- layer 1 ----
    gemm_wmma_f32<<<gemmBlkN, TB, 0, stream>>>(x, W1n2e, fA, N);
    hipMemsetAsync(fE, 0, (size_t)NE * DIM * sizeof(float), stream);
    scatter_kernel<<<scatGrid, TB, 0, stream>>>(fA, fE, nidx, eidx, DeI, NNZ);
    bias_prelu_kernel<<<cdiv((long long)NE * DIM, TB), TB, 0, stream>>>(fE, b1n2e, pa, (size_t)NE * DIM);
    gemm_wmma_f32<<<gemmBlkE, TB, 0, stream>>>(fE, W1e2n, fA, NE);
    hipMemsetAsync(fB, 0, (size_t)N * DIM * sizeof(float), stream);
    scatter_kernel<<<scatGrid, TB, 0, stream>>>(fA, fB, eidx, nidx, DnI, NNZ);
    bias_prelu_kernel<<<cdiv((long long)N * DIM, TB), TB, 0, stream>>>(fB, b1e2n, pa, (size_t)N * DIM);

    // ---- layer 2 ----
    gemm_wmma_f32<<<gemmBlkN, TB, 0, stream>>>(fB, W2n2e, fA, N);
    hipMemsetAsync(fE, 0, (size_t)NE * DIM * sizeof(float), stream);
    scatter_kernel<<<scatGrid, TB, 0, stream>>>(fA, fE, nidx, eidx, DeI, NNZ);
    bias_prelu_kernel<<<cdiv((long long)NE * DIM, TB), TB, 0, stream>>>(fE, b2n2e, pa, (size_t)NE * DIM);
    gemm_wmma_f32<<<gemmBlkE, TB, 0, stream>>>(fE, W2e2n, fA, NE);
    hipMemsetAsync(fB, 0, (size_t)N * DIM * sizeof(float), stream);
    scatter_kernel<<<scatGrid, TB, 0, stream>>>(fA, fB, eidx, nidx, DnI, NNZ);
    bias_prelu_kernel<<<cdiv((long long)N * DIM, TB), TB, 0, stream>>>(fB, b2e2n, pa, (size_t)N * DIM);

    // ---- query pooling + prediction ----
    hipMemsetAsync(Ze1, 0, (size_t)NQ * DIM * sizeof(float), stream);
    hipMemsetAsync(Ze2, 0, (size_t)NQ * DIM * sizeof(float), stream);
    hipMemsetAsync(c1,  0, (size_t)NQ * sizeof(float), stream);
    hipMemsetAsync(c2,  0, (size_t)NQ * sizeof(float), stream);
    qscatter_kernel<<<qscatGrid, TB, 0, stream>>>(fB, Vmap1, Emap1, Ze1, c1, LQ);
    qscatter_kernel<<<qscatGrid, TB, 0, stream>>>(fB, Vmap2, Emap2, Ze2, c2, LQ);
    pred_kernel<<<cdiv((long long)NQ * 32, TB), TB, 0, stream>>>(Ze1, c1, Ze2, c2, out, NQ);
}